// AlignTTSLoss_7481833029808
// MI455X (gfx1250) — compile-verified
//
#include <hip/hip_runtime.h>
#include <hip/hip_bf16.h>

// ---------------------------------------------------------------------------
// AlignTTS loss, fused for MI455X (gfx1250, wave32).
//   blk [0,64)            : MDN forward log-sum-exp scan (1 WG per batch)
//   blk [64,4160)         : fused SSIM (vertical conv in LDS, horizontal conv
//                           as banded GEMM via v_wmma_f32_16x16x4_f32,
//                           SSIM epilogue + reduction in registers)
//   blk [4160,6208)       : masked-MSE reduction (spec)
//   blk [6208,6224)       : masked-MSE reduction (dur)
// ---------------------------------------------------------------------------

typedef __attribute__((ext_vector_type(2))) float v2f;
typedef __attribute__((ext_vector_type(8))) float v8f;

#define NEGV    (-10000.0f)
#define SSIM_C1 (0.0001f)    // 0.01^2
#define SSIM_C2 (0.0009f)    // 0.03^2
#define GEXPC   (-0.222222222f)   // -1/(2*sigma^2) = -1/4.5
#define GINVSUM (0.26601174f)     // 1 / sum(exp(-d^2/4.5), d=-5..5)

#define B_    64
#define TSEQ  256
#define TMEL  1024
#define DD    80

#define NB_MDN  64
#define NB_SSIM (64 * 64)    // 64 batches x 64 row-blocks of 16
#define NB_SPEC 2048
#define NB_DUR  16
#define NB_TOTAL (NB_MDN + NB_SSIM + NB_SPEC + NB_DUR)

// ws float slots
#define WS_SPEC 0
#define WS_DUR  1
#define WS_SSIM 2
#define WS_MDN  3

// vq row stride in dwords: 100 mod 64 == 36 == 4*9 -> the 64 dwords of an
// A-frag ds_load (lanes x 2 dwords, halves offset by +2) tile all 64 LDS
// banks bijectively: bank = 4*(9*m mod 16) + j.  Zero conflicts.
#define VQS 100

// shared memory layout (bytes)
//  SSIM : xt[26][80] @0 (8320)  yt @8320 (8320)  vq[5][16][VQS] @16640 (32000)
//         gauss[11] @48640 (44)  red[8] @48688 (32)  -> 48720
//  MDN  : alpha[256] @0 (1024)  tile[256][33] @1024 (33792) -> 34816
#define SMEM_BYTES 48736

__device__ __forceinline__ float waveSum(float v) {
#pragma unroll
  for (int off = 16; off > 0; off >>= 1) v += __shfl_xor(v, off, 32);
  return v;
}

__global__ __launch_bounds__(256) void aligntts_fused_kernel(
    const float* __restrict__ logp,      // (64,256,1024)
    const float* __restrict__ dec_out,   // (64,1024,80)
    const float* __restrict__ dec_tgt,   // (64,1024,80)
    const int*   __restrict__ dec_lens,  // (64,)
    const float* __restrict__ dur_out,   // (64,256)
    const float* __restrict__ dur_tgt,   // (64,256)
    const int*   __restrict__ in_lens,   // (64,)
    float* __restrict__ ws)
{
  __shared__ __align__(16) char smem[SMEM_BYTES];
  const int tid = threadIdx.x;
  const int blk = blockIdx.x;

  if (blk < NB_MDN) {
    // ================= MDN forward scan =================
    float* alpha = (float*)smem;               // [256]
    float* tile  = (float*)(smem + 1024);      // [256][33] padded stride
    const int b   = blk;
    const int mel = dec_lens[b];
    const int txt = in_lens[b];
    const int s   = tid;
    const float* lpb = logp + (size_t)b * TSEQ * TMEL;

    float a0 = (s == 0) ? lpb[0] : NEGV;
    alpha[s] = a0;
    if (mel == 1 && s == txt - 1) atomicAdd(&ws[WS_MDN], a0);
    __syncthreads();

    for (int t0 = 0; t0 < TMEL; t0 += 32) {
      // stage logp[:, t0..t0+31] into LDS, coalesced (32 lanes span one row)
      for (int i = tid; i < TSEQ * 32; i += 256) {
        int s2 = i >> 5, j = i & 31;
        tile[s2 * 33 + j] = lpb[(s2 << 10) + t0 + j];
        if (t0 + 32 < TMEL)
          __builtin_prefetch(&lpb[(s2 << 10) + t0 + 32 + j], 0, 3);
      }
      __syncthreads();
      const int tts = (t0 == 0) ? 1 : 0;
      for (int tt = tts; tt < 32; ++tt) {
        const int t = t0 + tt;
        float lp = tile[s * 33 + tt];
        float ap = alpha[s] + 1e-4f;
        float sh = ((s > 0) ? alpha[s - 1] : NEGV) + 1e-4f;
        float mx = fmaxf(ap, sh), mn = fminf(ap, sh);
        // short serial chain: v_exp + v_log; __expf underflows cleanly for
        // mn-mx << 0, and log(1+e) vs log1p differs by < 1e-7 absolute here.
        float nv = mx + __logf(1.0f + __expf(mn - mx)) + lp;
        __syncthreads();
        alpha[s] = nv;
        if (t == mel - 1 && s == txt - 1) atomicAdd(&ws[WS_MDN], nv);
        __syncthreads();
      }
    }
  } else if (blk < NB_MDN + NB_SSIM) {
    // ================= fused SSIM tile =================
    float* xt = (float*)smem;                  // [26][80] (16 rows + 2*5 halo)
    float* yt = (float*)(smem + 8320);
    float* vq = (float*)(smem + 16640);        // [5][16][VQS], data at col+8
    float* ga = (float*)(smem + 48640);        // [11] gaussian (vertical pass)
    float* red = (float*)(smem + 48688);       // [8]

    const int sb = blk - NB_MDN;
    const int b  = sb >> 6;
    const int h0 = (sb & 63) << 4;             // 16 output rows per block
    const int dlen = dec_lens[b];

    if (tid == 0) {
      float raw[11]; float sum = 0.f;
      for (int i = 0; i < 11; ++i) {
        float d = (float)(i - 5);
        raw[i] = expf(d * d * GEXPC);
        sum += raw[i];
      }
      for (int i = 0; i < 11; ++i) ga[i] = raw[i] / sum;
    }

    // ---- stage 1: load masked x,y tile with vertical halo ----
    const float* xs = dec_out + (size_t)b * TMEL * DD;
    const float* ysr = dec_tgt + (size_t)b * TMEL * DD;
    for (int i = tid; i < 26 * 80; i += 256) {
      int rr = i / 80, c = i - rr * 80;
      int h = h0 + rr - 5;
      bool ok = (h >= 0) & (h < TMEL) & (h < dlen);
      xt[rr * 80 + c] = ok ? xs[h * 80 + c] : 0.f;
      yt[rr * 80 + c] = ok ? ysr[h * 80 + c] : 0.f;
    }
    // zero apron of vq: padded cols [0,8) and [88,96) (k in [-8,0) u [80,88))
    for (int i = tid; i < 5 * 16 * 16; i += 256) {
      int row = i >> 4, cc = i & 15;
      int col = (cc < 8) ? cc : (cc + 80);
      vq[row * VQS + col] = 0.f;
    }
    __syncthreads();

    // ---- stage 2: vertical 11-tap conv of {x, y, xx, yy, xy} ----
    for (int e = tid; e < 5 * 16 * 80; e += 256) {
      int q = e / 1280;
      int rc = e - q * 1280;
      int r = rc / 80, c = rc - r * 80;
      float acc = 0.f;
#pragma unroll
      for (int k = 0; k < 11; ++k) {
        float xv = xt[(r + k) * 80 + c];
        float yv = yt[(r + k) * 80 + c];
        float val = (q == 0) ? xv
                  : (q == 1) ? yv
                  : (q == 2) ? xv * xv
                  : (q == 3) ? yv * yv
                  :            xv * yv;
        acc = fmaf(ga[k], val, acc);
      }
      vq[(q * 16 + r) * VQS + 8 + c] = acc;
    }
    __syncthreads();

    // ---- stage 3: horizontal conv = banded GEMM via WMMA fp32 ----
    const int w = tid >> 5;
    const int lane = tid & 31;
    if (w < 5) {                               // waves 0..4 own col-tiles n0=w
      const int n0 = w;
      const int m = lane & 15;                 // A row / D col index
      const int half = lane >> 4;
      const int nn = n0 * 16 + m;              // output column for B frag

      v8f acc[5];
#pragma unroll
      for (int q = 0; q < 5; ++q)
#pragma unroll
        for (int j = 0; j < 8; ++j) acc[q][j] = 0.f;

      // band k in [16*n0-8, 16*n0+23]: constant 8-trip K-tile loop; the zero
      // apron in vq makes out-of-range K contribute exactly 0 (zero padding).
#pragma unroll
      for (int u = 0; u < 8; ++u) {
        const int kt = 4 * n0 - 2 + u;
        const int k0 = kt * 4 + half * 2;      // lanes 0-15: K=4kt+{0,1}; 16-31: +2
        // B[k][n] = g[n-k+5], fast-exp (args tiny; ~1e-6 rel err is plenty)
        int dd0 = nn - k0;
        int dd1 = dd0 - 1;
        float w0 = __expf((float)(dd0 * dd0) * GEXPC) * GINVSUM;
        float w1 = __expf((float)(dd1 * dd1) * GEXPC) * GINVSUM;
        v2f bf;
        bf.x = (dd0 * dd0 <= 25) ? w0 : 0.f;
        bf.y = (dd1 * dd1 <= 25) ? w1 : 0.f;
#pragma unroll
        for (int q = 0; q < 5; ++q) {
          const float* vb = vq + (q * 16 + m) * VQS + 8;
          v2f af; af.x = vb[k0]; af.y = vb[k0 + 1];
          acc[q] = __builtin_amdgcn_wmma_f32_16x16x4_f32(
              false, af, false, bf, (short)0, acc[q], false, false);
        }
      }

      // ---- SSIM epilogue on accumulators, D VGPR j <-> (M=j+8*half, N=nn)
      float wsum = 0.f;
#pragma unroll
      for (int j = 0; j < 8; ++j) {
        float mu1 = acc[0][j], mu2 = acc[1][j];
        float cxx = acc[2][j], cyy = acc[3][j], cxy = acc[4][j];
        float mu1s = mu1 * mu1, mu2s = mu2 * mu2, mu12 = mu1 * mu2;
        float s1 = cxx - mu1s, s2 = cyy - mu2s, s12 = cxy - mu12;
        float num = (2.f * mu12 + SSIM_C1) * (2.f * s12 + SSIM_C2);
        float den = (mu1s + mu2s + SSIM_C1) * (s1 + s2 + SSIM_C2);
        wsum += num / den;
      }
      wsum = waveSum(wsum);
      if (lane == 0) red[w] = wsum;
    }
    __syncthreads();
    if (tid == 0)
      atomicAdd(&ws[WS_SSIM], red[0] + red[1] + red[2] + red[3] + red[4]);
  } else if (blk < NB_MDN + NB_SSIM + NB_SPEC) {
    // ================= spec masked MSE =================
    float* red = (float*)smem;
    const float4* xo = (const float4*)dec_out;
    const float4* xg = (const float4*)dec_tgt;
    const int N4 = B_ * TMEL * (DD / 4);
    int gid = (blk - NB_MDN - NB_SSIM) * 256 + tid;
    float sum = 0.f;
    for (int i = gid; i < N4; i += NB_SPEC * 256) {
      int e = i * 4;
      int b = e / (TMEL * DD);
      int h = (e / DD) & (TMEL - 1);
      if (h < dec_lens[b]) {
        float4 a = xo[i], t = xg[i];
        float d0 = a.x - t.x, d1 = a.y - t.y, d2 = a.z - t.z, d3 = a.w - t.w;
        sum += d0 * d0 + d1 * d1 + d2 * d2 + d3 * d3;
      }
    }
    sum = waveSum(sum);
    int w = tid >> 5, lane = tid & 31;
    if (lane == 0) red[w] = sum;
    __syncthreads();
    if (tid == 0) {
      float t = 0.f;
      for (int k = 0; k < 8; ++k) t += red[k];
      atomicAdd(&ws[WS_SPEC], t);
    }
  } else {
    // ================= dur masked MSE =================
    float* red = (float*)smem;
    int gid = (blk - NB_MDN - NB_SSIM - NB_SPEC) * 256 + tid;
    float sum = 0.f;
    for (int i = gid; i < B_ * TSEQ; i += NB_DUR * 256) {
      int b = i >> 8, s = i & 255;
      if (s < in_lens[b]) {
        float d = dur_out[i] - dur_tgt[i];
        sum += d * d;
      }
    }
    sum = waveSum(sum);
    int w = tid >> 5, lane = tid & 31;
    if (lane == 0) red[w] = sum;
    __syncthreads();
    if (tid == 0) {
      float t = 0.f;
      for (int k = 0; k < 8; ++k) t += red[k];
      atomicAdd(&ws[WS_DUR], t);
    }
  }
}

__global__ void aligntts_finalize_kernel(const int* __restrict__ dec_lens,
                                         const int* __restrict__ in_lens,
                                         const float* __restrict__ ws,
                                         float* __restrict__ out)
{
  if (threadIdx.x == 0 && blockIdx.x == 0) {
    float sd = 0.f, si = 0.f;
    for (int b = 0; b < B_; ++b) { sd += (float)dec_lens[b]; si += (float)in_lens[b]; }
    float spec = ws[WS_SPEC] / (sd * (float)DD);
    float dur  = ws[WS_DUR] / si;
    float ssim = 1.f - ws[WS_SSIM] / (float)(B_ * TMEL * DD);
    float mdn  = -(ws[WS_MDN] / (float)B_) / (float)TSEQ;
    out[0] = spec + ssim + dur + mdn;
    out[1] = spec;
    out[2] = ssim;
    out[3] = dur;
    out[4] = mdn;
  }
}

extern "C" void kernel_launch(void* const* d_in, const int* in_sizes, int n_in,
                              void* d_out, int out_size, void* d_ws, size_t ws_size,
                              hipStream_t stream) {
  const float* logp    = (const float*)d_in[0];
  const float* dec_out = (const float*)d_in[1];
  const float* dec_tgt = (const float*)d_in[2];
  const int*   dec_len = (const int*)d_in[3];
  const float* dur_out = (const float*)d_in[4];
  const float* dur_tgt = (const float*)d_in[5];
  const int*   in_len  = (const int*)d_in[6];
  float* ws  = (float*)d_ws;
  float* out = (float*)d_out;

  hipMemsetAsync(ws, 0, 4 * sizeof(float), stream);
  aligntts_fused_kernel<<<NB_TOTAL, 256, 0, stream>>>(
      logp, dec_out, dec_tgt, dec_len, dur_out, dur_tgt, in_len, ws);
  aligntts_finalize_kernel<<<1, 32, 0, stream>>>(dec_len, in_len, ws, out);
}